// BaseEncLoss_15264313770474
// MI455X (gfx1250) — compile-verified
//
#include <hip/hip_runtime.h>
#include <hip/hip_bf16.h>

typedef float v2f __attribute__((ext_vector_type(2)));
typedef float v8f __attribute__((ext_vector_type(8)));

#define NB   16           // batch
#define NC   19           // classes
#define HH   512
#define WW   512
#define GG   16           // grid_size
#define HC   (HH / GG)    // 32 cell rows
#define WCC  (WW / GG)    // 32 cell cols
#define HT   1024
#define WT   1024

__global__ void zero_out_k(float* out) { out[0] = 0.0f; }

// ---------------- Phase 1: per-cell class-presence bitmask -----------------
// One block per (b, cy) cell-row. Covers label rows 2*(16*cy + r), r=0..15,
// even columns only (nearest-neighbor downsample). Each thread's 32 samples
// land in exactly 2 cells (w = t and w = 256 + t), so only 2 LDS atomicOr's.
__global__ __launch_bounds__(256) void presence_k(const int* __restrict__ targets,
                                                  unsigned int* __restrict__ presence)
{
    __shared__ unsigned int cellMask[WCC];
    const int t   = threadIdx.x;
    const int blk = blockIdx.x;          // b*HC + cy
    const int cy  = blk & (HC - 1);
    const int b   = blk >> 5;

    if (t < WCC) cellMask[t] = 0u;
    __syncthreads();

    unsigned int m0 = 0u, m1 = 0u;
#pragma unroll
    for (int i = 0; i < 32; ++i) {
        const int r16    = i >> 1;                 // 0..15 row within cell-row
        const int w      = ((i & 1) << 8) + t;     // 0..511
        const int srcRow = (cy * GG + r16) * 2;    // downsample rows
        const int srcCol = w * 2;                  // downsample cols
        const int label  = targets[((size_t)b * HT + srcRow) * WT + srcCol];
        const unsigned int bit = 1u << label;
        if (i & 1) m1 |= bit; else m0 |= bit;
    }
    atomicOr(&cellMask[t >> 4], m0);
    atomicOr(&cellMask[16 + (t >> 4)], m1);
    __syncthreads();

    if (t < WCC) presence[(size_t)blk * WCC + t] = cellMask[t];
}

// ---------------- Phase 2: streaming BCE + WMMA reduction ------------------
// One block per (b, c, cy): an 8192-float contiguous slab (16 rows x 512).
// Presence row (32 cells) cached in LDS as float {0,1}. Per-lane partials are
// kept as the 16x4 f32 A-fragment; one v_wmma_f32_16x16x4_f32 vs ones-B
// reduces all 64 lane-partials of the wave exactly in f32.
__global__ __launch_bounds__(256) void bce_k(const float* __restrict__ preds,
                                             const unsigned int* __restrict__ presence,
                                             float* __restrict__ out,
                                             const float scale)
{
    __shared__ float seRow[WCC];
    __shared__ float waveSum[8];

    const int t   = threadIdx.x;
    const int blk = blockIdx.x;          // ((b*NC + c)*HC + cy)
    const int cy  = blk % HC;
    const int bc  = blk / HC;
    const int c   = bc % NC;
    const int b   = bc / NC;

    if (t < WCC) {
        const unsigned int m = presence[((size_t)(b * HC + cy)) * WCC + t];
        seRow[t] = (float)((m >> c) & 1u);
    }
    __syncthreads();

    const float* base = preds + (((size_t)(b * NC + c)) * HH + (size_t)cy * GG) * WW;

    v2f acc; acc.x = 0.0f; acc.y = 0.0f;

#pragma unroll
    for (int i = 0; i < 8; ++i) {
        const int f = i * 256 + t;            // float4 index in [0, 2048)
        const int w = (f & 127) << 2;         // column 0..508 (stays in one cell)
        const float4 x4 = *reinterpret_cast<const float4*>(base + (size_t)f * 4);

        // prefetch the next slab's lines into L2 (speculative, drop-safe)
        __builtin_prefetch(base + (size_t)f * 4 + 8192, 0, 1);

        const float se = seRow[w >> 4];

        const float t0 = fmaxf(x4.x, 0.0f) - x4.x * se + __logf(1.0f + __expf(-fabsf(x4.x)));
        const float t1 = fmaxf(x4.y, 0.0f) - x4.y * se + __logf(1.0f + __expf(-fabsf(x4.y)));
        const float t2 = fmaxf(x4.z, 0.0f) - x4.z * se + __logf(1.0f + __expf(-fabsf(x4.z)));
        const float t3 = fmaxf(x4.w, 0.0f) - x4.w * se + __logf(1.0f + __expf(-fabsf(x4.w)));

        acc.x += t0 + t2;
        acc.y += t1 + t3;
    }

    // Cross-lane reduction via WMMA: D[m,n] = sum_k A[m,k] * 1
    v2f ones; ones.x = 1.0f; ones.y = 1.0f;
    v8f czero = {};
    v8f d = __builtin_amdgcn_wmma_f32_16x16x4_f32(
        /*neg_a=*/false, acc, /*neg_b=*/false, ones,
        /*c_mod=*/(short)0, czero, /*reuse_a=*/false, /*reuse_b=*/false);

    float s = d[0] + d[1] + d[2] + d[3] + d[4] + d[5] + d[6] + d[7];
    s += __shfl_xor(s, 16, 32);            // combine the two lane-halves

    const int lane = t & 31;
    const int wv   = t >> 5;
    if (lane == 0) waveSum[wv] = s;
    __syncthreads();

    if (t == 0) {
        float bs = 0.0f;
#pragma unroll
        for (int k = 0; k < 8; ++k) bs += waveSum[k];
        atomicAdd(out, bs * scale);
    }
}

extern "C" void kernel_launch(void* const* d_in, const int* in_sizes, int n_in,
                              void* d_out, int out_size, void* d_ws, size_t ws_size,
                              hipStream_t stream)
{
    const float* preds   = (const float*)d_in[0];
    const int*   targets = (const int*)d_in[1];
    // d_in[2] = grid_size (16), fixed by the reference shapes.

    unsigned int* presence = (unsigned int*)d_ws;   // 16384 * 4B = 64 KB
    float* out = (float*)d_out;

    zero_out_k<<<1, 1, 0, stream>>>(out);

    presence_k<<<NB * HC, 256, 0, stream>>>(targets, presence);

    const float scale = 1.0f / (float)((size_t)NB * NC * HH * WW);
    bce_k<<<NB * NC * HC, 256, 0, stream>>>(preds, presence, out, scale);
}